// KeypointLoss_78030965833906
// MI455X (gfx1250) — compile-verified
//
#include <hip/hip_runtime.h>
#include <stdint.h>

// ---------------------------------------------------------------------------
// KeypointLoss for MI455X (gfx1250).
// Memory-bound: ~277 MB of HBM traffic, AI ~0.37 FLOP/B -> ~12us floor at
// 23.3 TB/s. No WMMA (matrix engine useless here); instead we exercise the
// CDNA5 async data-mover path: global_load_async_to_lds_b128 with manual
// s_wait_asynccnt pipelining (4-deep, wave-private LDS staging) so each wave
// keeps 12 x 512B lines in flight through the async unit.
// Fully deterministic: no float atomics; per-block partials in d_ws.
// ---------------------------------------------------------------------------

#ifndef USE_ASYNC
#define USE_ASYNC 1
#endif

namespace {
constexpr int B_  = 32;
constexpr int S_  = 2;
constexpr int K_  = 11;
constexpr int C_  = 7;
constexpr int HW_ = 256 * 256;               // 65536
constexpr int THREADS = 256;                 // 8 waves (wave32)
constexpr int T_ITERS = HW_ / 4 / THREADS;   // 64 float4 iterations / thread
constexpr int WAVES   = THREADS / 32;
}  // namespace

#if USE_ASYNC
__device__ __forceinline__ void async_ld_b128(unsigned lds_byte_off, const void* gaddr) {
  // VDST = per-lane LDS byte address, VADDR = 64-bit global address, no saddr.
  asm volatile("global_load_async_to_lds_b128 %0, %1, off"
               :: "v"(lds_byte_off),
                  "v"((unsigned long long)(uintptr_t)gaddr)
               : "memory");
}
#endif

// One block per (b,k): streams HW elements once, producing
//   wsF[(b*K+k)*2 + s] = sum over HW of (pred[b,s,k,:] - hm[b,k,:])^2
//   wsI[b*K+k]         = argmax over HW of hm[b,k,:]   (first occurrence)
__global__ void __launch_bounds__(THREADS)
hm_loss_argmax(const float* __restrict__ preds,   // [B,S,K,HW]
               const float* __restrict__ hms,     // [B,K,HW]
               float* __restrict__ wsF,           // [B*K*2]
               int* __restrict__ wsI)             // [B*K]
{
  const int bk  = blockIdx.x;
  const int b   = bk / K_;
  const int k   = bk - b * K_;
  const int tid = threadIdx.x;

  const float4* gh = reinterpret_cast<const float4*>(hms   + (size_t)bk * HW_) + tid;
  const float4* g0 = reinterpret_cast<const float4*>(preds + ((size_t)(b * S_ + 0) * K_ + k) * HW_) + tid;
  const float4* g1 = reinterpret_cast<const float4*>(preds + ((size_t)(b * S_ + 1) * K_ + k) * HW_) + tid;

  float s0 = 0.f, s1 = 0.f;
  float bestv = -__builtin_inff();
  int   besti = 0;

  // 48 KB: wave-private staging, 4 buffers x 3 streams x 32 lanes x 16B.
  __shared__ float4 smem4[WAVES * 4 * 3 * 32];

  auto consume = [&](int t, float4 h, float4 a, float4 c) {
    float d;
    d = a.x - h.x; s0 = fmaf(d, d, s0);
    d = a.y - h.y; s0 = fmaf(d, d, s0);
    d = a.z - h.z; s0 = fmaf(d, d, s0);
    d = a.w - h.w; s0 = fmaf(d, d, s0);
    d = c.x - h.x; s1 = fmaf(d, d, s1);
    d = c.y - h.y; s1 = fmaf(d, d, s1);
    d = c.z - h.z; s1 = fmaf(d, d, s1);
    d = c.w - h.w; s1 = fmaf(d, d, s1);
    const int fi = (t * THREADS + tid) * 4;   // ascending per thread -> strict '>' keeps first
    if (h.x > bestv) { bestv = h.x; besti = fi;     }
    if (h.y > bestv) { bestv = h.y; besti = fi + 1; }
    if (h.z > bestv) { bestv = h.z; besti = fi + 2; }
    if (h.w > bestv) { bestv = h.w; besti = fi + 3; }
  };

#if USE_ASYNC
  const int wave = tid >> 5;
  const int lane = tid & 31;
  const unsigned lds_base = (unsigned)(uintptr_t)(&smem4[0]);

  auto slot = [&](int buf, int stream) -> unsigned {   // float4-slot index
    return (unsigned)(((wave * 4 + buf) * 3 + stream) * 32 + lane);
  };
  auto stage = [&](int t) {
    const int buf = t & 3;
    async_ld_b128(lds_base + slot(buf, 0) * 16u, gh + (size_t)t * THREADS);
    async_ld_b128(lds_base + slot(buf, 1) * 16u, g0 + (size_t)t * THREADS);
    async_ld_b128(lds_base + slot(buf, 2) * 16u, g1 + (size_t)t * THREADS);
  };

  stage(0); stage(1); stage(2);
  for (int t = 0; t < T_ITERS; ++t) {
    if (t + 3 < T_ITERS) {
      // Close WAR window: LDS reads of the buffer we are about to overwrite
      // (consumed at iteration t-1) must have completed.
      asm volatile("s_wait_dscnt 0" ::: "memory");
      stage(t + 3);
      asm volatile("s_wait_asynccnt 9" ::: "memory");   // stage t's 3 ops done
    } else if (t + 3 == T_ITERS) {
      asm volatile("s_wait_asynccnt 6" ::: "memory");
    } else if (t + 2 == T_ITERS) {
      asm volatile("s_wait_asynccnt 3" ::: "memory");
    } else {
      asm volatile("s_wait_asynccnt 0" ::: "memory");
    }
    const int buf = t & 3;
    float4 h = smem4[slot(buf, 0)];
    float4 a = smem4[slot(buf, 1)];
    float4 c = smem4[slot(buf, 2)];
    consume(t, h, a, c);
  }
  asm volatile("s_wait_asynccnt 0" ::: "memory");
#else
  #pragma unroll 4
  for (int t = 0; t < T_ITERS; ++t) {
    float4 h = gh[(size_t)t * THREADS];
    float4 a = g0[(size_t)t * THREADS];
    float4 c = g1[(size_t)t * THREADS];
    consume(t, h, a, c);
  }
#endif

  // ---- block reduction (reuse staging LDS) ----
  __syncthreads();
  float* rs0 = reinterpret_cast<float*>(smem4);
  float* rs1 = rs0 + THREADS;
  float* rmv = rs1 + THREADS;
  int*   rmi = reinterpret_cast<int*>(rmv + THREADS);
  rs0[tid] = s0; rs1[tid] = s1; rmv[tid] = bestv; rmi[tid] = besti;
  __syncthreads();
  for (int off = THREADS / 2; off > 0; off >>= 1) {
    if (tid < off) {
      rs0[tid] += rs0[tid + off];
      rs1[tid] += rs1[tid + off];
      const float v2 = rmv[tid + off];
      const int   i2 = rmi[tid + off];
      if (v2 > rmv[tid] || (v2 == rmv[tid] && i2 < rmi[tid])) {
        rmv[tid] = v2; rmi[tid] = i2;
      }
    }
    __syncthreads();
  }
  if (tid == 0) {
    wsF[bk * 2 + 0] = rs0[0];
    wsF[bk * 2 + 1] = rs1[0];
    wsI[bk]         = rmi[0];
  }
}

// One block per (b,s): label loss over 77 gathered terms + finalize hm loss.
__global__ void __launch_bounds__(128)
labels_finalize(const float* __restrict__ lb,      // [B,S,C,HW]
                const float* __restrict__ labels,  // [B,K,C]
                const float* __restrict__ wsF,     // [B*K*2]
                const int* __restrict__ wsI,       // [B*K]
                float* __restrict__ out)           // [128] = combined(64) ++ labels(64)
{
  const int bs = blockIdx.x;     // b*S + s
  const int b  = bs >> 1;
  const int s  = bs & 1;
  const int t  = threadIdx.x;

  float v = 0.f;
  if (t < K_ * C_) {
    const int k   = t / C_;
    const int c   = t - k * C_;
    const int idx = wsI[b * K_ + k];
    const float g = lb[(((size_t)(b * S_ + s)) * C_ + c) * HW_ + idx];
    const float l = labels[((size_t)b * K_ + k) * C_ + c];
    const float d = g - l;
    v = d * d;
  }
  __shared__ float red[128];
  red[t] = v;
  __syncthreads();
  for (int off = 64; off > 0; off >>= 1) {
    if (t < off) red[t] += red[t + off];
    __syncthreads();
  }
  if (t == 0) {
    out[B_ * S_ + bs] = red[0] * (1.0f / (float)(K_ * C_));
    float hs = 0.f;
    #pragma unroll
    for (int k = 0; k < K_; ++k) hs += wsF[(b * K_ + k) * 2 + s];
    out[bs] = hs * (1.0f / (float)((size_t)K_ * HW_));
  }
}

extern "C" void kernel_launch(void* const* d_in, const int* in_sizes, int n_in,
                              void* d_out, int out_size, void* d_ws, size_t ws_size,
                              hipStream_t stream) {
  (void)in_sizes; (void)n_in; (void)out_size; (void)ws_size;
  const float* hm_preds = (const float*)d_in[0];   // [32,2,11,256,256]
  const float* lb_preds = (const float*)d_in[1];   // [32,2,7,256,256]
  const float* heatmaps = (const float*)d_in[2];   // [32,11,256,256]
  const float* labels   = (const float*)d_in[3];   // [32,11,7]
  float* out = (float*)d_out;

  float* wsF = (float*)d_ws;                       // 704 floats
  int*   wsI = (int*)((char*)d_ws + 4096);         // 352 ints

  hm_loss_argmax<<<B_ * K_, THREADS, 0, stream>>>(hm_preds, heatmaps, wsF, wsI);
  labels_finalize<<<B_ * S_, 128, 0, stream>>>(lb_preds, labels, wsF, wsI, out);
}